// CombinedLoss_54554674594330
// MI455X (gfx1250) — compile-verified
//
#include <hip/hip_runtime.h>

typedef __attribute__((ext_vector_type(2))) float v2f;
typedef __attribute__((ext_vector_type(8))) float v8f;

#define ALPHA 0.9f
#define BETA  0.1f
#define GAMMA 0.1f
#define EPSV  1e-8f

// d_ws layout (floats): ws[0] = sum (y-l)^2, ws[1] = sum cos, ws[2] = full-matrix hinge sum

__global__ void cl_init_ws(float* ws) {
    if (threadIdx.x < 3) ws[threadIdx.x] = 0.0f;
}

__global__ void __launch_bounds__(256)
cl_vector_losses(const float* __restrict__ y, const float* __restrict__ l,
                 float* __restrict__ ws, int N) {
    __shared__ float s_sq[256];
    __shared__ float s_cs[256];
    int i = blockIdx.x * 256 + threadIdx.x;
    float sq = 0.0f, cs = 0.0f;
    if (i < N) {
        float yv = y[i], lv = l[i];
        float e = yv - lv;
        sq = e * e;
        cs = (yv * lv) / (fmaxf(fabsf(yv), EPSV) * fmaxf(fabsf(lv), EPSV));
    }
    s_sq[threadIdx.x] = sq;
    s_cs[threadIdx.x] = cs;
    __syncthreads();
    for (int off = 128; off > 0; off >>= 1) {
        if (threadIdx.x < off) {
            s_sq[threadIdx.x] += s_sq[threadIdx.x + off];
            s_cs[threadIdx.x] += s_cs[threadIdx.x + off];
        }
        __syncthreads();
    }
    if (threadIdx.x == 0) {
        atomicAdd(&ws[0], s_sq[0]);
        atomicAdd(&ws[1], s_cs[0]);
    }
}

// Pairwise hinge: each wave owns 16 rows, loops over all 16-wide column tiles.
// dy and dl 16x16 tiles are formed as rank-2 outer products via
// V_WMMA_F32_16X16X4_F32:  A = [y_i | 1 | 0 | 0] (16x4),  B = [1 ; -y_j ; 0 ; 0] (4x16).
// hinge = max(0, -sign(dl)*dy); full-matrix sum == 2 * upper-triangle sum.
__global__ void __launch_bounds__(256)
cl_pairwise_hinge(const float* __restrict__ y, const float* __restrict__ l,
                  float* __restrict__ ws, int N) {
    const int tid  = threadIdx.x;
    const int lane = tid & 31;
    const int wave = tid >> 5;                       // 0..7
    const int rowBase = (blockIdx.x * 8 + wave) * 16;
    const int m = lane & 15;
    const bool lo = (lane < 16);                     // lanes 0..15 carry K=0,1; 16..31 carry zero pads
    const float onelo = lo ? 1.0f : 0.0f;

    // A operands (loop-invariant). Unconditional loads from valid addresses; mask via cndmask.
    float yiAll = y[rowBase + m];
    float liAll = l[rowBase + m];
    v2f aY, aL;
    aY.x = lo ? yiAll : 0.0f;  aY.y = onelo;
    aL.x = lo ? liAll : 0.0f;  aL.y = onelo;

    v8f acc = {};   // per-lane hinge accumulator (8 tile elements per lane)
#pragma unroll 4
    for (int col = 0; col < N; col += 16) {
        float yjAll = y[col + m];
        float ljAll = l[col + m];
        v2f bY, bL;
        bY.x = onelo;  bY.y = lo ? -yjAll : 0.0f;
        bL.x = onelo;  bL.y = lo ? -ljAll : 0.0f;

        v8f zc = {};
        // dy[M][N] = y_i[M] - y_j[N],  dl[M][N] = l_i[M] - l_j[N]
        v8f dy = __builtin_amdgcn_wmma_f32_16x16x4_f32(
            false, aY, false, bY, (short)0, zc, false, false);
        v8f dl = __builtin_amdgcn_wmma_f32_16x16x4_f32(
            false, aL, false, bL, (short)0, zc, false, false);

#pragma unroll
        for (int r = 0; r < 8; ++r) {
            float d = dy[r];
            float e = dl[r];
            // Branchless: hinge = max(0, -sign(e)*d), exact dl==0 handling.
            float t = (e > 0.0f) ? -d : d;   // v_cndmask
            t = (e == 0.0f) ? 0.0f : t;      // v_cndmask
            acc[r] += fmaxf(t, 0.0f);        // v_max + v_add
        }
    }

    float partial = ((acc[0] + acc[1]) + (acc[2] + acc[3]))
                  + ((acc[4] + acc[5]) + (acc[6] + acc[7]));

    __shared__ float red[256];
    red[tid] = partial;
    __syncthreads();
    for (int off = 128; off > 0; off >>= 1) {
        if (tid < off) red[tid] += red[tid + off];
        __syncthreads();
    }
    if (tid == 0) atomicAdd(&ws[2], red[0]);
}

__global__ void cl_finalize(const float* __restrict__ ws, float* __restrict__ out, int N) {
    if (threadIdx.x == 0 && blockIdx.x == 0) {
        float nf  = (float)N;
        float mse = ws[0] / nf;
        float sim = 1.0f - ws[1] / nf;
        // upper-triangle sum = full sum / 2; divided by N(N-1)/2  ->  full / (N*(N-1))
        float mrl = ws[2] / (nf * (nf - 1.0f));
        out[0] = ALPHA * mse + BETA * mrl + GAMMA * sim;
    }
}

extern "C" void kernel_launch(void* const* d_in, const int* in_sizes, int n_in,
                              void* d_out, int out_size, void* d_ws, size_t ws_size,
                              hipStream_t stream) {
    const float* y = (const float*)d_in[0];
    const float* l = (const float*)d_in[1];
    float* out = (float*)d_out;
    float* ws  = (float*)d_ws;
    const int N = in_sizes[0];   // 8192; multiple of 128 assumed

    cl_init_ws<<<1, 32, 0, stream>>>(ws);
    cl_vector_losses<<<(N + 255) / 256, 256, 0, stream>>>(y, l, ws, N);
    cl_pairwise_hinge<<<N / 128, 256, 0, stream>>>(y, l, ws, N);  // 8 waves * 16 rows per WG
    cl_finalize<<<1, 1, 0, stream>>>(ws, out, N);
}